// unEmbedding_27986006901451
// MI455X (gfx1250) — compile-verified
//
#include <hip/hip_runtime.h>

// out[b,n] = dot(x[b,n,:], W[n,:]) + bias[n]
// x: [256, 1024, 512] f32, W: [1024, 512] f32, bias: [1024] f32, out: [256*1024] f32
//
// HBM-bound (512 MB of x @ 23.3 TB/s ~= 22 us). Strategy:
//  - one wave32 per 16 consecutive output rows (same b, 16 consecutive neurons)
//  - lane L -> row (L & 15); half-wave (L >> 4) splits each 32-float d-chunk
//    so every iteration reads one full 128B line per row with b128 loads
//  - per-lane fp32 partial sums; final cross-lane reduction done with a single
//    v_wmma_f32_16x16x4_f32 against an all-ones B matrix (A-layout puts row M's
//    K-slots in lanes M and M+16, so D[m,*] = full row sum, already in C layout)

#define D_MODEL     512
#define NUM_NEURONS 1024
#define BATCH       256

typedef float v2f __attribute__((ext_vector_type(2)));
typedef float v4f __attribute__((ext_vector_type(4)));
typedef float v8f __attribute__((ext_vector_type(8)));

__global__ __launch_bounds__(256) void neuron_dot_wmma_kernel(
    const float* __restrict__ x,
    const float* __restrict__ W,
    const float* __restrict__ bias,
    float* __restrict__ out) {
  const int lane = threadIdx.x & 31;
  const int wave = threadIdx.x >> 5;                 // 8 waves per block
  const int tile = blockIdx.x * 8 + wave;            // 16 rows per tile
  const int r0   = tile << 4;                        // first flat row (b*N + n)
  const int m    = lane & 15;                        // row within tile
  const int half = lane >> 4;                        // which 16-float sub-chunk

  const int row = r0 + m;                            // this lane's flat row
  const int n   = row & (NUM_NEURONS - 1);           // neuron index (N = 1024)

  const float* __restrict__ xrow = x + (size_t)row * D_MODEL + half * 16;
  const float* __restrict__ wrow = W + (size_t)n   * D_MODEL + half * 16;

  float s0 = 0.0f, s1 = 0.0f;

  // Each iteration: this lane covers 16 floats of the row; the paired lane
  // (m <-> m+16) covers the other 16 => one full 128B cacheline per row.
  for (int d = 0; d < D_MODEL; d += 32) {
    const float* xp = xrow + d;
    const float* wp = wrow + d;
    v4f xa = *(const v4f*)(xp + 0);
    v4f xb = *(const v4f*)(xp + 4);
    v4f xc = *(const v4f*)(xp + 8);
    v4f xd = *(const v4f*)(xp + 12);
    v4f wa = *(const v4f*)(wp + 0);
    v4f wb = *(const v4f*)(wp + 4);
    v4f wc = *(const v4f*)(wp + 8);
    v4f wd = *(const v4f*)(wp + 12);

    s0 = fmaf(xa.x, wa.x, s0);
    s0 = fmaf(xa.y, wa.y, s0);
    s0 = fmaf(xa.z, wa.z, s0);
    s0 = fmaf(xa.w, wa.w, s0);
    s0 = fmaf(xb.x, wb.x, s0);
    s0 = fmaf(xb.y, wb.y, s0);
    s0 = fmaf(xb.z, wb.z, s0);
    s0 = fmaf(xb.w, wb.w, s0);
    s1 = fmaf(xc.x, wc.x, s1);
    s1 = fmaf(xc.y, wc.y, s1);
    s1 = fmaf(xc.z, wc.z, s1);
    s1 = fmaf(xc.w, wc.w, s1);
    s1 = fmaf(xd.x, wd.x, s1);
    s1 = fmaf(xd.y, wd.y, s1);
    s1 = fmaf(xd.z, wd.z, s1);
    s1 = fmaf(xd.w, wd.w, s1);
  }

  // WMMA tail reduction. fp32 A-matrix 16x4 layout:
  //   lane M    : VGPR0 = A[M, K=0], VGPR1 = A[M, K=1]
  //   lane M+16 : VGPR0 = A[M, K=2], VGPR1 = A[M, K=3]
  // With B = all ones: D[m, j] = s0(m) + s1(m) + s0(m+16) + s1(m+16) = row sum.
  v2f a;   a.x = s0;   a.y = s1;
  v2f one; one.x = 1.0f; one.y = 1.0f;
  v8f c = {};
  c = __builtin_amdgcn_wmma_f32_16x16x4_f32(
      /*neg_a=*/false, a, /*neg_b=*/false, one,
      /*c_mod=*/(short)0, c, /*reuse_a=*/false, /*reuse_b=*/false);

  // C/D layout: c[j] holds D[j, lane] (lanes 0-15) and D[8+j, lane-16]
  // (lanes 16-31); all columns of a row are equal, so lane 0 owns rows 0-7
  // and lane 16 owns rows 8-15. Emit each half as two b128 stores with bias.
  if ((lane & 15) == 0) {
    const int rbase = r0 + half * 8;
    const int nbase = (r0 & (NUM_NEURONS - 1)) + half * 8;
    v4f b0 = *(const v4f*)(bias + nbase);
    v4f b1 = *(const v4f*)(bias + nbase + 4);
    v4f o0, o1;
    o0.x = c[0] + b0.x;  o0.y = c[1] + b0.y;
    o0.z = c[2] + b0.z;  o0.w = c[3] + b0.w;
    o1.x = c[4] + b1.x;  o1.y = c[5] + b1.y;
    o1.z = c[6] + b1.z;  o1.w = c[7] + b1.w;
    *(v4f*)(out + rbase)     = o0;
    *(v4f*)(out + rbase + 4) = o1;
  }
}

extern "C" void kernel_launch(void* const* d_in, const int* in_sizes, int n_in,
                              void* d_out, int out_size, void* d_ws, size_t ws_size,
                              hipStream_t stream) {
  (void)in_sizes; (void)n_in; (void)out_size; (void)d_ws; (void)ws_size;
  const float* x    = (const float*)d_in[0];
  const float* W    = (const float*)d_in[1];
  const float* bias = (const float*)d_in[2];
  float* out        = (float*)d_out;

  // 256*1024 rows / 16 rows per wave = 16384 waves; 8 waves (256 thr) per block.
  const int total_tiles = (BATCH * NUM_NEURONS) / 16;   // 16384
  const int blocks = total_tiles / 8;                   // 2048
  hipLaunchKernelGGL(neuron_dot_wmma_kernel, dim3(blocks), dim3(256), 0, stream,
                     x, W, bias, out);
}